// LabEmbedding_35983236006185
// MI455X (gfx1250) — compile-verified
//
#include <hip/hip_runtime.h>
#include <stdint.h>

// LabEmbedding: memory-bound f32 streaming kernel for MI455X (gfx1250).
// out[n,f] = s_v[n]*W[n,f] + s_w[n]*b[n,f], where s_v/s_w come from a
// time-weighted reduction over T=256 samples.
// CDNA5 features used: wave32 shuffle reduction, async global->LDS staging
// (GLOBAL_LOAD_ASYNC_TO_LDS_B128 + s_wait_asynccnt) to overlap the 6KB W/b
// row fetch with the reduction phase.

#define D_FF  768
#define T_LEN 256

__global__ __launch_bounds__(256) void lab_embedding_gfx1250_kernel(
    const float* __restrict__ times,
    const float* __restrict__ values,
    const float* __restrict__ W,
    const float* __restrict__ Bias,
    float* __restrict__ out)
{
    const int n   = blockIdx.x;    // variable/row index
    const int tid = threadIdx.x;   // 0..255

    __shared__ float sWb[2 * D_FF];   // [0..767] = W row, [768..1535] = bias row
    __shared__ float partials[16];    // 8 waves x {sum_t, sum_tv}
    __shared__ float bc[2];           // broadcast {s_v, s_w}

    // ---------------- Phase 0: async-stage W and bias rows into LDS --------
    // 768 floats per row = 192 lanes x 16B. Issued first so the fetch overlaps
    // the reduction below. Tracked by ASYNCcnt, not LOADcnt.
    if (tid < (D_FF / 4)) {
        const unsigned lds_base = (unsigned)(size_t)(&sWb[0]); // LDS offset (low 32 bits of flat addr)
        const unsigned lds_w = lds_base + (unsigned)(tid * 16);
        const unsigned lds_b = lds_base + (unsigned)(D_FF * 4 + tid * 16);
        const unsigned voff  = (unsigned)(tid * 16);           // byte offset within row
        const float* wrow = W    + (size_t)n * D_FF;
        const float* brow = Bias + (size_t)n * D_FF;
        // GVS form: LDS destination address from the vdst VGPR; global address is the SGPR base plus the 32-bit VGPR offset
        asm volatile("global_load_async_to_lds_b128 %0, %1, %2"
                     :: "v"(lds_w), "v"(voff), "s"(wrow) : "memory");
        asm volatile("global_load_async_to_lds_b128 %0, %1, %2"
                     :: "v"(lds_b), "v"(voff), "s"(brow) : "memory");
    }

    // ---------------- Phase 1: weighted reduction over T --------------------
    const float t = times [(size_t)n * T_LEN + tid];
    const float v = values[(size_t)n * T_LEN + tid];
    float x = t;        // partial sum of times
    float y = t * v;    // partial sum of times*values

    // wave32 butterfly reduction
    #pragma unroll
    for (int m = 16; m > 0; m >>= 1) {
        x += __shfl_xor(x, m, 32);
        y += __shfl_xor(y, m, 32);
    }
    if ((tid & 31) == 0) {
        partials[(tid >> 5) * 2 + 0] = x;
        partials[(tid >> 5) * 2 + 1] = y;
    }
    __syncthreads();

    if (tid == 0) {
        float sx = 0.0f, sy = 0.0f;
        #pragma unroll
        for (int w = 0; w < 8; ++w) { sx += partials[2 * w]; sy += partials[2 * w + 1]; }
        const bool zero = (sx == 0.0f);
        bc[0] = zero ? 0.0f : (sy / sx);  // s_v
        bc[1] = zero ? 0.0f : 1.0f;       // s_w
    }

    // Wait for this wave's async LDS writes; barrier makes all waves' visible.
    asm volatile("s_wait_asynccnt 0x0" ::: "memory");
    __syncthreads();

    // ---------------- Phase 2: stream the output row ------------------------
    const float s_v = bc[0];
    const float s_w = bc[1];
    if (tid < (D_FF / 4)) {
        const float4 w4 = *(const float4*)&sWb[tid * 4];
        const float4 b4 = *(const float4*)&sWb[D_FF + tid * 4];
        float4 o;
        o.x = fmaf(s_v, w4.x, s_w * b4.x);
        o.y = fmaf(s_v, w4.y, s_w * b4.y);
        o.z = fmaf(s_v, w4.z, s_w * b4.z);
        o.w = fmaf(s_v, w4.w, s_w * b4.w);
        *(float4*)(out + (size_t)n * D_FF + tid * 4) = o;
    }
}

extern "C" void kernel_launch(void* const* d_in, const int* in_sizes, int n_in,
                              void* d_out, int out_size, void* d_ws, size_t ws_size,
                              hipStream_t stream)
{
    const float* times  = (const float*)d_in[0];  // [N, T]
    const float* values = (const float*)d_in[1];  // [N, T]
    const float* W      = (const float*)d_in[2];  // [N, d_ff]
    const float* Bias   = (const float*)d_in[3];  // [N, d_ff]
    float* out = (float*)d_out;                   // [N, d_ff] (reshape is a no-op)

    const int N = in_sizes[2] / D_FF;             // 16384
    lab_embedding_gfx1250_kernel<<<N, 256, 0, stream>>>(times, values, W, Bias, out);
}